// Decoder_22789096472706
// MI455X (gfx1250) — compile-verified
//
#include <hip/hip_runtime.h>
#include <math.h>

// ---------------------------------------------------------------------------
// GCN 2-layer forward for MI455X (gfx1250, wave32).
// Dense transforms via V_WMMA_F32_16X16X4_F32 (fully unrolled, templated K);
// edge aggregation via L2-resident float4 gathers + global_atomic_add_f32.
// ---------------------------------------------------------------------------

typedef __attribute__((ext_vector_type(2))) float v2f;
typedef __attribute__((ext_vector_type(8))) float v8f;

// ---------------- degree / normalization ----------------

__global__ void deg_init_kernel(float* __restrict__ deg, int n) {
  int i = blockIdx.x * blockDim.x + threadIdx.x;
  if (i < n) deg[i] = 1.0f;  // self-loop contributes 1 to every node's degree
}

__global__ void deg_count_kernel(const int* __restrict__ dst,
                                 float* __restrict__ deg, int e) {
  int tid = blockIdx.x * blockDim.x + threadIdx.x;
  int stride = gridDim.x * blockDim.x;
  for (int i = tid; i < e; i += stride) {
    int d = __builtin_nontemporal_load(dst + i);
    atomicAdd(&deg[d], 1.0f);
  }
}

__global__ void dinv_kernel(float* __restrict__ deg, int n) {
  int i = blockIdx.x * blockDim.x + threadIdx.x;
  if (i < n) deg[i] = rsqrtf(deg[i]);  // deg >= 1 always (self-loops)
}

// ---------------- WMMA dense transform ----------------
// H[N,16] = X[N,KDIM] @ W[KDIM,WCOLS] zero-padded to 16 output columns.
// One wave per 16-row tile. F32 WMMA 16x16x4 layout:
//   A: lane m=lane&15 holds row m; VGPR0/1 = K {0,1} (lanes 0-15) or {2,3} (16-31)
//   B: lane col=lane&15 holds col; VGPR0/1 = K {0,1} / {2,3} mirrored
//   C/D: lane n=lane&15, VGPR v -> row v + 8*(lane>=16)
template <int KDIM, int WCOLS>
__global__ void gcn_gemm_wmma_kernel(const float* __restrict__ X,
                                     const float* __restrict__ W,
                                     float* __restrict__ H, int n) {
  int wave = (int)((blockIdx.x * blockDim.x + threadIdx.x) >> 5);
  int lane = (int)(threadIdx.x & 31);
  int ntiles = (n + 15) >> 4;
  if (wave >= ntiles) return;  // wave-uniform: EXEC stays all-1s for WMMA

  int hi  = lane >> 4;               // half-wave select
  int m   = lane & 15;               // A row within tile / B-D column
  int row = (wave << 4) + m;
  int rowc = row < n ? row : n - 1;  // clamp tail loads (store is guarded)
  int kb  = hi << 1;                 // K sub-offset: 0 or 2

  const float* xrow = X + (size_t)rowc * KDIM + kb;
  const int cc = (WCOLS < 16) ? (m < WCOLS ? m : 0) : m;
  const bool cok = (WCOLS < 16) ? (m < WCOLS) : true;

  v8f c = (v8f)0.0f;
#pragma unroll
  for (int k0 = 0; k0 < KDIM; k0 += 4) {
    v2f a, b;
    a.x = xrow[k0 + 0];
    a.y = xrow[k0 + 1];
    float b0 = W[(size_t)(k0 + kb + 0) * WCOLS + cc];
    float b1 = W[(size_t)(k0 + kb + 1) * WCOLS + cc];
    b.x = cok ? b0 : 0.0f;  // zero-pad columns WCOLS..15
    b.y = cok ? b1 : 0.0f;
    c = __builtin_amdgcn_wmma_f32_16x16x4_f32(
        /*neg_a=*/false, a, /*neg_b=*/false, b,
        /*c_mod=*/(short)0, c, /*reuse_a=*/false, /*reuse_b=*/false);
  }

  int base = wave << 4;
  if (base + 16 <= n) {
    // Full tile (common case, wave-uniform branch): unconditional stores,
    // one row pointer + immediate offsets -> clause-able store_b32 burst.
    float* hrow = H + (size_t)(base + (hi << 3)) * 16 + m;
#pragma unroll
    for (int v = 0; v < 8; ++v) hrow[(size_t)v * 16] = c[v];
  } else {
    // Partial tail tile (at most one per launch): per-row guard.
#pragma unroll
    for (int v = 0; v < 8; ++v) {
      int r = base + v + (hi << 3);
      if (r < n) H[(size_t)r * 16 + m] = c[v];
    }
  }
}

// ---------------- edge gather/scale/scatter-add ----------------
// h has row stride 16 floats (16B-aligned rows -> float4 gathers).
// agg has row stride F (16 for layer 1, 12 for layer 2 -> d_out).
template <int F>
__global__ void edge_scatter_kernel(const int* __restrict__ src,
                                    const int* __restrict__ dst,
                                    const float* __restrict__ dinv,
                                    const float* __restrict__ h,
                                    float* __restrict__ agg, int e) {
  int tid = blockIdx.x * blockDim.x + threadIdx.x;
  int stride = gridDim.x * blockDim.x;
  for (int i = tid; i < e; i += stride) {
    int s = __builtin_nontemporal_load(src + i);
    int d = __builtin_nontemporal_load(dst + i);
    float nrm = dinv[s] * dinv[d];
    const float4* hp = (const float4*)(h + (size_t)s * 16);
    float* ap = agg + (size_t)d * F;
#pragma unroll
    for (int q = 0; q < F / 4; ++q) {
      float4 v = hp[q];
      atomicAdd(ap + 4 * q + 0, v.x * nrm);
      atomicAdd(ap + 4 * q + 1, v.y * nrm);
      atomicAdd(ap + 4 * q + 2, v.z * nrm);
      atomicAdd(ap + 4 * q + 3, v.w * nrm);
    }
  }
}

// ---------------- epilogues (fold self-loop term, bias, activation) --------

__global__ void finalize1_kernel(float* __restrict__ agg,   // [N,16] in/out
                                 const float* __restrict__ h1,
                                 const float* __restrict__ dinv,
                                 const float* __restrict__ b1, int n) {
  int i = blockIdx.x * blockDim.x + threadIdx.x;
  if (i >= n) return;
  float di = dinv[i];
  float di2 = di * di;
  float4* ag4 = (float4*)(agg + (size_t)i * 16);
  const float4* h4 = (const float4*)(h1 + (size_t)i * 16);
  const float4* bb = (const float4*)b1;
#pragma unroll
  for (int q = 0; q < 4; ++q) {
    float4 a = ag4[q], h = h4[q], b = bb[q];
    a.x = fmaxf(a.x + h.x * di2 + b.x, 0.0f);
    a.y = fmaxf(a.y + h.y * di2 + b.y, 0.0f);
    a.z = fmaxf(a.z + h.z * di2 + b.z, 0.0f);
    a.w = fmaxf(a.w + h.w * di2 + b.w, 0.0f);
    ag4[q] = a;
  }
}

__device__ __forceinline__ float sigmoidf_(float v) {
  return 1.0f / (1.0f + __expf(-v));
}

__global__ void finalize2_kernel(float* __restrict__ out,       // [N,12] in/out
                                 const float* __restrict__ h2,  // [N,16] padded
                                 const float* __restrict__ dinv,
                                 const float* __restrict__ b2, int n) {
  int i = blockIdx.x * blockDim.x + threadIdx.x;
  if (i >= n) return;
  float di = dinv[i];
  float di2 = di * di;
  // out rows are 12 floats = 48B -> float4 aligned; h2 rows are 64B.
  float4* o4 = (float4*)(out + (size_t)i * 12);
  const float4* h4 = (const float4*)(h2 + (size_t)i * 16);
  const float4* bb = (const float4*)b2;
#pragma unroll
  for (int q = 0; q < 3; ++q) {
    float4 o = o4[q], h = h4[q], b = bb[q];
    o.x = sigmoidf_(o.x + h.x * di2 + b.x);
    o.y = sigmoidf_(o.y + h.y * di2 + b.y);
    o.z = sigmoidf_(o.z + h.z * di2 + b.z);
    o.w = sigmoidf_(o.w + h.w * di2 + b.w);
    o4[q] = o;
  }
}

// ---------------------------------------------------------------------------

extern "C" void kernel_launch(void* const* d_in, const int* in_sizes, int n_in,
                              void* d_out, int out_size, void* d_ws, size_t ws_size,
                              hipStream_t stream) {
  const float* x  = (const float*)d_in[0];   // [N, 8]
  const int*   ei = (const int*)d_in[1];     // [2, E] row-major
  const float* W1 = (const float*)d_in[2];   // [8, 16]
  const float* b1 = (const float*)d_in[3];   // [16]
  const float* W2 = (const float*)d_in[4];   // [16, 12]
  const float* b2 = (const float*)d_in[5];   // [12]
  float* out = (float*)d_out;                // [N, 12]

  const int N = in_sizes[0] / 8;
  const int E = in_sizes[1] / 2;
  const int* src = ei;
  const int* dst = ei + E;

  // Workspace layout (256B aligned slices)
  size_t off = 0;
  auto carve = [&](size_t bytes) {
    size_t o = off;
    off = (off + bytes + 255) & ~(size_t)255;
    return o;
  };
  char* ws = (char*)d_ws;
  float* dinv = (float*)(ws + carve((size_t)N * 4));        // deg -> dinv
  float* h1   = (float*)(ws + carve((size_t)N * 16 * 4));   // X @ W1
  float* agg1 = (float*)(ws + carve((size_t)N * 16 * 4));   // layer-1 agg -> relu(h)
  float* h2   = (float*)(ws + carve((size_t)N * 16 * 4));   // relu(h) @ W2 (padded)
  (void)ws_size;

  const int TB = 256;
  const int gridN = (N + TB - 1) / TB;
  int gridE = (E + TB - 1) / TB;
  if (gridE > 8192) gridE = 8192;  // grid-stride
  const int nTiles = (N + 15) / 16;
  const int gridW = (nTiles * 32 + TB - 1) / TB;  // 8 waves / block

  // Zero accumulation targets
  hipMemsetAsync(agg1, 0, (size_t)N * 16 * 4, stream);
  hipMemsetAsync(out, 0, (size_t)out_size * 4, stream);

  // Degree / normalization
  deg_init_kernel<<<gridN, TB, 0, stream>>>(dinv, N);
  deg_count_kernel<<<gridE, TB, 0, stream>>>(dst, dinv, E);
  dinv_kernel<<<gridN, TB, 0, stream>>>(dinv, N);

  // Layer 1: h1 = X @ W1 (8 -> 16), aggregate, relu
  gcn_gemm_wmma_kernel<8, 16><<<gridW, TB, 0, stream>>>(x, W1, h1, N);
  edge_scatter_kernel<16><<<gridE, TB, 0, stream>>>(src, dst, dinv, h1, agg1, E);
  finalize1_kernel<<<gridN, TB, 0, stream>>>(agg1, h1, dinv, b1, N);

  // Layer 2: h2 = relu(h) @ W2 (16 -> 12, padded to 16 cols), aggregate, sigmoid
  gcn_gemm_wmma_kernel<16, 12><<<gridW, TB, 0, stream>>>(agg1, W2, h2, N);
  edge_scatter_kernel<12><<<gridE, TB, 0, stream>>>(src, dst, dinv, h2, out, E);
  finalize2_kernel<<<gridN, TB, 0, stream>>>(out, h2, dinv, b2, N);
}